// DownBlock_4028679324199
// MI455X (gfx1250) — compile-verified
//
#include <hip/hip_runtime.h>

// CDNA5 (gfx1250) wave32 WMMA fragment types:
//  A (16x4 f32)  = 2 VGPRs/lane -> float ext-vector of 2
//  C/D (16x16 f32) = 8 VGPRs/lane -> float ext-vector of 8
typedef __attribute__((ext_vector_type(2))) float v2f;
typedef __attribute__((ext_vector_type(8))) float v8f;

// D = A(16x4) * B(4x16) + C, full fp32 matrix pipe.
// 8-arg form: (neg_a, A, neg_b, B, c_mod, C, reuse_a, reuse_b)
#define WMMA_F32(a, b, c) \
  __builtin_amdgcn_wmma_f32_16x16x4_f32(false, (a), false, (b), (short)0, (c), false, false)

// Problem constants (from setup_inputs)
constexpr int Bn  = 16;
constexpr int Cin = 128;
constexpr int Fch = 64;
constexpr int Hh  = 128;
constexpr int Ww  = 128;
constexpr int HW  = Hh * Ww;      // 16384
constexpr int KK  = 6;            // downConv kernel
constexpr int Kflat = Fch * KK * KK; // 2304 (im2col K)
constexpr int OH  = 64;
constexpr int OW  = 64;

// ---------------------------------------------------------------------------
// Kernel 1: fused pac-1x1-conv (guide kernel == exp(0) == 1 exactly) +
//           bias + PReLU.  Pure GEMM: h[b,f,p] = sum_c w[f,c] * x[b,c,p].
// Block = 256 threads = 8 waves. Weights (64x128 f32 = 32 KB) staged in LDS.
// Each wave owns 16 pixels and computes all 64 output channels with 4
// accumulators so x is read from global exactly once.
// ---------------------------------------------------------------------------
__global__ __launch_bounds__(256)
void pac1x1_prelu_kernel(const float* __restrict__ x,
                         const float* __restrict__ pw,   // [64,128]
                         const float* __restrict__ pb,   // [64]
                         const float* __restrict__ alpha1,
                         float* __restrict__ h)          // [16,64,128,128]
{
    __shared__ float sw[Fch * Cin];                       // 32 KB
    const int tid = threadIdx.x;
    for (int i = tid; i < Fch * Cin; i += 256) sw[i] = pw[i];
    __syncthreads();

    const int blk   = blockIdx.x;        // 16 batches * 128 pixel tiles
    const int b     = blk >> 7;
    const int ptile = blk & 127;
    const int wave  = tid >> 5;
    const int lane  = tid & 31;
    const int ln    = lane & 15;
    const int hi    = lane >> 4;
    const int p0    = ptile * 128 + wave * 16;

    const float* xb = x + (size_t)b * Cin * HW + p0 + ln;
    const float a1  = alpha1[0];

    v8f acc0 = {}, acc1 = {}, acc2 = {}, acc3 = {};

    for (int c0 = 0; c0 < Cin; c0 += 4) {
        const int cA = c0 + 2 * hi;       // K rows this half-wave supplies
        v2f bv;
        bv.x = xb[(size_t)cA * HW];       // B[k=cA  ][n=ln]
        bv.y = xb[(size_t)(cA + 1) * HW]; // B[k=cA+1][n=ln]

        v2f a0 = *(const v2f*)&sw[(0 * 16 + ln) * Cin + cA];
        v2f a1v = *(const v2f*)&sw[(1 * 16 + ln) * Cin + cA];
        v2f a2v = *(const v2f*)&sw[(2 * 16 + ln) * Cin + cA];
        v2f a3v = *(const v2f*)&sw[(3 * 16 + ln) * Cin + cA];

        acc0 = WMMA_F32(a0,  bv, acc0);
        acc1 = WMMA_F32(a1v, bv, acc1);
        acc2 = WMMA_F32(a2v, bv, acc2);
        acc3 = WMMA_F32(a3v, bv, acc3);
    }

    float* hb = h + (size_t)b * Fch * HW + p0 + ln;
#pragma unroll
    for (int t = 0; t < 4; ++t) {
        v8f acc = (t == 0) ? acc0 : (t == 1) ? acc1 : (t == 2) ? acc2 : acc3;
#pragma unroll
        for (int v = 0; v < 8; ++v) {
            const int f = t * 16 + v + 8 * hi;    // C/D layout: row = v + 8*hi
            float val = acc[v] + pb[f];
            val = (val >= 0.f) ? val : a1 * val;  // PReLU
            hb[(size_t)f * HW] = val;
        }
    }
}

// ---------------------------------------------------------------------------
// Kernel 2: 6x6 stride-2 pad-2 conv as im2col GEMM (K = 64*36 = 2304) +
//           bias + PReLU. Block = 128 threads = 4 waves, one (b, oy, 16-ox)
//           tile. The 64ch x 6row x 36col input halo (54 KB) is staged in LDS
//           with zero padding so the K-loop has no bounds checks (EXEC stays
//           all-ones for WMMA). Each wave handles 16 output channels.
// ---------------------------------------------------------------------------
__global__ __launch_bounds__(128)
void conv6x6s2_prelu_kernel(const float* __restrict__ h,   // [16,64,128,128]
                            const float* __restrict__ cw,  // [64,64,6,6]
                            const float* __restrict__ cb,  // [64]
                            const float* __restrict__ alpha2,
                            float* __restrict__ out)       // [16,64,64,64]
{
    __shared__ float sh[Fch * 6 * 36];                     // 13824 f = 54 KB
    const int tid = threadIdx.x;
    const int oxt = blockIdx.x;       // 0..3
    const int oy  = blockIdx.y;       // 0..63
    const int b   = blockIdx.z;       // 0..15
    const int ox0 = oxt * 16;
    const int iy0 = 2 * oy  - 2;
    const int ix0 = 2 * ox0 - 2;

    const float* hb = h + (size_t)b * Fch * HW;
    // 13824 / 128 = 108 iterations per thread, fully uniform, coalesced in j.
    for (int i = tid; i < Fch * 6 * 36; i += 128) {
        const int c   = i / 216;
        const int rem = i - c * 216;
        const int r   = rem / 36;
        const int j   = rem - r * 36;
        const int iy  = iy0 + r;
        const int ix  = ix0 + j;
        float v = 0.f;
        if ((unsigned)iy < (unsigned)Hh && (unsigned)ix < (unsigned)Ww)
            v = hb[(size_t)c * HW + iy * Ww + ix];
        sh[i] = v;
    }
    __syncthreads();

    const int wave = tid >> 5;
    const int lane = tid & 31;
    const int ln   = lane & 15;
    const int hi   = lane >> 4;
    const int f0   = wave * 16;
    const float a2 = alpha2[0];

    v8f acc = {};
    const float* wrow = cw + (size_t)(f0 + ln) * Kflat;   // contiguous in k

    for (int c = 0; c < Fch; ++c) {
        const float* shc = sh + c * 216;
#pragma unroll
        for (int s = 0; s < 36; s += 4) {
            const int kA = s + 2 * hi;                     // 0..34 (+1 below)
            v2f a = *(const v2f*)&wrow[c * 36 + kA];       // A[f0+ln][kA..kA+1]

            // B[k][n] = h[c, iy0 + k/6, ix0 + 2*n + k%6] from padded LDS tile
            const int ry1 = kA / 6,        kx1 = kA - 6 * ry1;
            const int k2  = kA + 1;
            const int ry2 = k2 / 6,        kx2 = k2 - 6 * ry2;
            v2f bv;
            bv.x = shc[ry1 * 36 + kx1 + 2 * ln];
            bv.y = shc[ry2 * 36 + kx2 + 2 * ln];

            acc = WMMA_F32(a, bv, acc);
        }
    }

    const size_t obase = ((size_t)b * Fch) * (OH * OW) + (size_t)oy * OW + ox0 + ln;
#pragma unroll
    for (int v = 0; v < 8; ++v) {
        const int f = f0 + v + 8 * hi;
        float val = acc[v] + cb[f];
        val = (val >= 0.f) ? val : a2 * val;               // PReLU
        out[obase + (size_t)f * (OH * OW)] = val;
    }
}

// ---------------------------------------------------------------------------
extern "C" void kernel_launch(void* const* d_in, const int* in_sizes, int n_in,
                              void* d_out, int out_size, void* d_ws, size_t ws_size,
                              hipStream_t stream) {
    const float* x      = (const float*)d_in[0];
    // d_in[1] = guide: mathematically unused (adaptive kernel == exp(0) == 1)
    const float* pac_w  = (const float*)d_in[2];   // [64,128,1,1]
    const float* pac_b  = (const float*)d_in[3];   // [64]
    const float* alpha1 = (const float*)d_in[4];   // [1]
    const float* alpha2 = (const float*)d_in[5];   // [1]
    const float* conv_w = (const float*)d_in[6];   // [64,64,6,6]
    const float* conv_b = (const float*)d_in[7];   // [64]
    float* out = (float*)d_out;                    // [16,64,64,64]
    float* h   = (float*)d_ws;                     // [16,64,128,128] = 64 MB

    // Stage 1: 1x1 conv GEMM + bias + PReLU  -> h (workspace)
    pac1x1_prelu_kernel<<<dim3(Bn * 128), dim3(256), 0, stream>>>(
        x, pac_w, pac_b, alpha1, h);

    // Stage 2: 6x6 s2 conv (im2col WMMA) + bias + PReLU -> out
    conv6x6s2_prelu_kernel<<<dim3(4, OH, Bn), dim3(128), 0, stream>>>(
        h, conv_w, conv_b, alpha2, out);
}